// PointNetFeaturePropagation_33655363732078
// MI455X (gfx1250) — compile-verified
//
#include <hip/hip_runtime.h>
#include <hip/hip_bf16.h>

// ---------------------------------------------------------------------------
// PointNet Feature Propagation for MI455X (gfx1250, wave32, WMMA).
// Pipeline:
//   1. transpose points_sa  [B,128,N]  -> x[:, 0:128]     (LDS tiled)
//   2. transpose points_now [B,256,S]  -> pnT [B,S,256]   (LDS tiled)
//   3. 3-NN over S=2048 (xyz_now staged in LDS), weights = inv-dist normalized
//   4. gather+blend 3 rows of pnT -> x[:, 128:384]
//   5. GEMM0 (65536x384 @ 384x256) via V_WMMA_F32_16X16X4_F32, +bias
//   6. per-channel sum/sumsq reduction (float atomics) -> BN+ReLU in place
//   7. GEMM1 (65536x256 @ 256x128) via WMMA, +bias
//   8. reduction -> BN+ReLU fused with transpose to out [B,128,N]
// ---------------------------------------------------------------------------

typedef __attribute__((ext_vector_type(2))) float v2f;
typedef __attribute__((ext_vector_type(8))) float v8f;

static constexpr int B_  = 8;
static constexpr int N_  = 8192;
static constexpr int S_  = 2048;
static constexpr int D1_ = 128;
static constexpr int D2_ = 256;
static constexpr int C0_ = D1_ + D2_;   // 384
static constexpr int H0_ = 256;
static constexpr int H1_ = 128;
static constexpr long M_ = (long)B_ * N_;  // 65536 rows
static constexpr float BN_EPS = 1e-5f;

// ---------------------------------------------------------------------------
// Zero the BN statistics accumulators (ws is not re-poisoned between replays).
// ---------------------------------------------------------------------------
__global__ void zero_stats_kernel(float* stats) {
    for (int i = threadIdx.x; i < 768; i += blockDim.x) stats[i] = 0.0f;
}

// ---------------------------------------------------------------------------
// Tiled transpose: in [B, C, NN] channel-major  ->  out rows of length
// out_stride at column offset col_off (row index = b*NN + n).
// grid: (NN/32, C/32, B), block: (32, 8)
// ---------------------------------------------------------------------------
__global__ __launch_bounds__(256)
void transpose_kernel(const float* __restrict__ in, float* __restrict__ out,
                      int C, int NN, int out_stride, int col_off) {
    __shared__ float tile[32][33];
    const int b  = blockIdx.z;
    const int n0 = blockIdx.x * 32;
    const int c0 = blockIdx.y * 32;
    #pragma unroll
    for (int i = threadIdx.y; i < 32; i += 8)
        tile[i][threadIdx.x] = in[((long)b * C + c0 + i) * NN + n0 + threadIdx.x];
    __syncthreads();
    #pragma unroll
    for (int i = threadIdx.y; i < 32; i += 8)
        out[((long)b * NN + n0 + i) * out_stride + col_off + c0 + threadIdx.x] =
            tile[threadIdx.x][i];
}

// ---------------------------------------------------------------------------
// 3-nearest-neighbor search. One thread per query point; xyz_now for the
// batch is staged in LDS (3 * 2048 * 4B = 24 KB, well inside 320 KB/WGP).
// grid: B * N/256, block: 256
// ---------------------------------------------------------------------------
__global__ __launch_bounds__(256)
void knn3_kernel(const float* __restrict__ xyz_sa, const float* __restrict__ xyz_now,
                 int* __restrict__ idx_out, float* __restrict__ w_out) {
    __shared__ float sx[S_], sy[S_], sz[S_];
    const int blocks_per_b = N_ / 256;
    const int b     = blockIdx.x / blocks_per_b;
    const int ntile = blockIdx.x % blocks_per_b;
    const float* now = xyz_now + (long)b * 3 * S_;
    for (int i = threadIdx.x; i < S_; i += 256) {
        sx[i] = now[i];
        sy[i] = now[S_ + i];
        sz[i] = now[2 * S_ + i];
    }
    __syncthreads();

    const int n = ntile * 256 + threadIdx.x;
    const float* sa = xyz_sa + (long)b * 3 * N_;
    const float px = sa[n], py = sa[N_ + n], pz = sa[2 * N_ + n];

    float d0 = 1e30f, d1 = 1e30f, d2 = 1e30f;
    int   i0 = 0,     i1 = 0,     i2 = 0;
    for (int s = 0; s < S_; ++s) {
        const float dx = px - sx[s];
        const float dy = py - sy[s];
        const float dz = pz - sz[s];
        const float d  = dx * dx + dy * dy + dz * dz;
        if (d < d2) {
            if (d < d1) {
                if (d < d0) { d2 = d1; i2 = i1; d1 = d0; i1 = i0; d0 = d; i0 = s; }
                else        { d2 = d1; i2 = i1; d1 = d;  i1 = s; }
            } else          { d2 = d;  i2 = s; }
        }
    }
    const float r0 = 1.0f / (d0 + 1e-8f);
    const float r1 = 1.0f / (d1 + 1e-8f);
    const float r2 = 1.0f / (d2 + 1e-8f);
    const float inv = 1.0f / (r0 + r1 + r2);
    const long p = (long)b * N_ + n;
    idx_out[p * 3 + 0] = i0;  w_out[p * 3 + 0] = r0 * inv;
    idx_out[p * 3 + 1] = i1;  w_out[p * 3 + 1] = r1 * inv;
    idx_out[p * 3 + 2] = i2;  w_out[p * 3 + 2] = r2 * inv;
}

// ---------------------------------------------------------------------------
// Weighted gather of 3 rows of pnT [B,S,256] -> x[:, 128:384].
// grid: B*N blocks, block: 256 (one thread per feature channel, coalesced)
// ---------------------------------------------------------------------------
__global__ __launch_bounds__(256)
void gather_interp_kernel(const float* __restrict__ pnT, const int* __restrict__ idx,
                          const float* __restrict__ w, float* __restrict__ x) {
    const long p = blockIdx.x;
    const int  c = threadIdx.x;
    const int  b = (int)(p / N_);
    const int* ip = idx + p * 3;
    const float* wp = w + p * 3;
    const float* base = pnT + (long)b * S_ * D2_;
    const float v = wp[0] * base[(long)ip[0] * D2_ + c]
                  + wp[1] * base[(long)ip[1] * D2_ + c]
                  + wp[2] * base[(long)ip[2] * D2_ + c];
    x[p * C0_ + D1_ + c] = v;
}

// ---------------------------------------------------------------------------
// fp32 WMMA GEMM:  Y[M, NOUT] = A[M, K] * W[NOUT, K]^T + bias
// Each wave computes a 16x64 tile: A fragment reused across 4 B fragments.
// Per-lane fragment layout matches V_WMMA_F32_16X16X4_F32 (ISA 7.12.2):
//   A 16x4 : lanes 0-15 hold (M=lane, K=k0,k0+1); lanes 16-31 hold K=k0+2,k0+3
//   B 4x16 : mirrored with N in place of M
//   C/D    : VGPR r -> row r (+8 for upper lane half), col = lane&15
// grid: (M/16)*(NOUT/64)/8 blocks, block: 256 (8 waves), all wave-uniform flow.
// ---------------------------------------------------------------------------
template <int K, int NOUT>
__global__ __launch_bounds__(256)
void gemm_wmma_kernel(const float* __restrict__ A, const float* __restrict__ W,
                      const float* __restrict__ bias, float* __restrict__ Y) {
    const int lane  = threadIdx.x & 31;
    const int wave  = threadIdx.x >> 5;
    const int gwave = blockIdx.x * 8 + wave;
    constexpr int NT64 = NOUT / 64;
    const int mbase = (gwave / NT64) * 16;
    const int nbase = (gwave % NT64) * 64;
    const int lhalf = lane >> 4;      // 0 or 1
    const int lmod  = lane & 15;

    const long aoff = (long)(mbase + lmod) * K + 2 * lhalf;
    long boff[4];
    #pragma unroll
    for (int t = 0; t < 4; ++t)
        boff[t] = (long)(nbase + t * 16 + lmod) * K + 2 * lhalf;

    v8f acc[4] = {};
    for (int k0 = 0; k0 < K; k0 += 4) {
        const v2f a  = *(const v2f*)(A + aoff + k0);
        const v2f b0 = *(const v2f*)(W + boff[0] + k0);
        const v2f b1 = *(const v2f*)(W + boff[1] + k0);
        const v2f b2 = *(const v2f*)(W + boff[2] + k0);
        const v2f b3 = *(const v2f*)(W + boff[3] + k0);
        acc[0] = __builtin_amdgcn_wmma_f32_16x16x4_f32(false, a, false, b0,
                                                       (short)0, acc[0], false, false);
        acc[1] = __builtin_amdgcn_wmma_f32_16x16x4_f32(false, a, false, b1,
                                                       (short)0, acc[1], false, false);
        acc[2] = __builtin_amdgcn_wmma_f32_16x16x4_f32(false, a, false, b2,
                                                       (short)0, acc[2], false, false);
        acc[3] = __builtin_amdgcn_wmma_f32_16x16x4_f32(false, a, false, b3,
                                                       (short)0, acc[3], false, false);
    }

    const int rowadd = 8 * lhalf;
    #pragma unroll
    for (int t = 0; t < 4; ++t) {
        const int col = nbase + t * 16 + lmod;
        const float bv = bias[col];
        #pragma unroll
        for (int r = 0; r < 8; ++r) {
            Y[(long)(mbase + r + rowadd) * NOUT + col] = acc[t][r] + bv;
        }
    }
}

// ---------------------------------------------------------------------------
// Per-channel sum / sum-of-squares over M rows (for training-mode BN).
// block: C threads (channel per thread, coalesced rows), grid-strided rows,
// float atomics to finalize. grid: 256
// ---------------------------------------------------------------------------
template <int C>
__global__ __launch_bounds__(C)
void reduce_stats_kernel(const float* __restrict__ Y, float* __restrict__ sum,
                         float* __restrict__ sumsq) {
    const int c = threadIdx.x;
    float s = 0.0f, q = 0.0f;
    for (long r = blockIdx.x; r < M_; r += gridDim.x) {
        const float v = Y[r * C + c];
        s += v;
        q += v * v;
    }
    atomicAdd(&sum[c], s);
    atomicAdd(&sumsq[c], q);
}

// ---------------------------------------------------------------------------
// BN (training stats) + ReLU, in place.
// ---------------------------------------------------------------------------
template <int C>
__global__ __launch_bounds__(256)
void bn_relu_kernel(float* __restrict__ Y, const float* __restrict__ sum,
                    const float* __restrict__ sumsq, const float* __restrict__ g,
                    const float* __restrict__ beta) {
    const long total = M_ * C;
    const float invM = 1.0f / (float)M_;
    for (long i = (long)blockIdx.x * blockDim.x + threadIdx.x; i < total;
         i += (long)gridDim.x * blockDim.x) {
        const int c = (int)(i % C);
        const float mean = sum[c] * invM;
        const float var  = sumsq[c] * invM - mean * mean;
        const float v = (Y[i] - mean) * rsqrtf(var + BN_EPS) * g[c] + beta[c];
        Y[i] = v > 0.0f ? v : 0.0f;
    }
}

// ---------------------------------------------------------------------------
// Final BN + ReLU fused with transpose: y1 [B*N, 128] -> out [B, 128, N].
// grid: (N/256, 128, B), block: 256 (coalesced writes over n)
// ---------------------------------------------------------------------------
__global__ __launch_bounds__(256)
void bn_relu_out_kernel(const float* __restrict__ Y1, const float* __restrict__ sum,
                        const float* __restrict__ sumsq, const float* __restrict__ g,
                        const float* __restrict__ beta, float* __restrict__ out) {
    const int n = blockIdx.x * 256 + threadIdx.x;
    const int c = blockIdx.y;
    const int b = blockIdx.z;
    const float invM = 1.0f / (float)M_;
    const float mean = sum[c] * invM;
    const float var  = sumsq[c] * invM - mean * mean;
    const float v = (Y1[((long)b * N_ + n) * H1_ + c] - mean)
                        * rsqrtf(var + BN_EPS) * g[c] + beta[c];
    out[((long)b * H1_ + c) * N_ + n] = v > 0.0f ? v : 0.0f;
}

// ---------------------------------------------------------------------------
// Host launcher
// ---------------------------------------------------------------------------
extern "C" void kernel_launch(void* const* d_in, const int* in_sizes, int n_in,
                              void* d_out, int out_size, void* d_ws, size_t ws_size,
                              hipStream_t stream) {
    const float* xyz_sa     = (const float*)d_in[0];
    const float* xyz_now    = (const float*)d_in[1];
    const float* points_sa  = (const float*)d_in[2];
    const float* points_now = (const float*)d_in[3];
    const float* W0    = (const float*)d_in[4];
    const float* b0    = (const float*)d_in[5];
    const float* g0    = (const float*)d_in[6];
    const float* beta0 = (const float*)d_in[7];
    const float* W1    = (const float*)d_in[8];
    const float* b1    = (const float*)d_in[9];
    const float* g1    = (const float*)d_in[10];
    const float* beta1 = (const float*)d_in[11];
    float* out = (float*)d_out;

    // Workspace layout (floats)
    float* ws    = (float*)d_ws;
    float* pnT   = ws;                                         // B*S*D2   = 4,194,304
    int*   kidx  = (int*)(ws + (long)B_ * S_ * D2_);           // B*N*3    =   196,608
    float* kw    = ws + (long)B_ * S_ * D2_ + M_ * 3;          // B*N*3    =   196,608
    float* stats = kw + M_ * 3;                                // 768
    float* sum0  = stats;        float* sq0 = stats + 256;
    float* sum1  = stats + 512;  float* sq1 = stats + 640;
    float* x     = stats + 768;                                // M*384 = 25,165,824
    float* y0    = x  + M_ * C0_;                              // M*256 = 16,777,216
    float* y1    = y0 + M_ * H0_;                              // M*128 =  8,388,608

    // 0) zero BN accumulators
    zero_stats_kernel<<<1, 256, 0, stream>>>(stats);

    // 1) transpose points_sa [B,128,N] -> x[:, 0:128]
    transpose_kernel<<<dim3(N_ / 32, D1_ / 32, B_), dim3(32, 8), 0, stream>>>(
        points_sa, x, D1_, N_, C0_, 0);

    // 2) transpose points_now [B,256,S] -> pnT [B,S,256]
    transpose_kernel<<<dim3(S_ / 32, D2_ / 32, B_), dim3(32, 8), 0, stream>>>(
        points_now, pnT, D2_, S_, D2_, 0);

    // 3) 3-NN search
    knn3_kernel<<<B_ * (N_ / 256), 256, 0, stream>>>(xyz_sa, xyz_now, kidx, kw);

    // 4) weighted gather -> x[:, 128:384]
    gather_interp_kernel<<<(int)M_, 256, 0, stream>>>(pnT, kidx, kw, x);

    // 5) GEMM0: y0 = x @ W0^T + b0   (M=65536, K=384, N=256)
    gemm_wmma_kernel<C0_, H0_><<<(int)((M_ / 16) * (H0_ / 64) / 8), 256, 0, stream>>>(
        x, W0, b0, y0);

    // 6) BN stats + BN/ReLU for layer 0
    reduce_stats_kernel<H0_><<<256, H0_, 0, stream>>>(y0, sum0, sq0);
    bn_relu_kernel<H0_><<<2048, 256, 0, stream>>>(y0, sum0, sq0, g0, beta0);

    // 7) GEMM1: y1 = y0 @ W1^T + b1  (M=65536, K=256, N=128)
    gemm_wmma_kernel<H0_, H1_><<<(int)((M_ / 16) * (H1_ / 64) / 8), 256, 0, stream>>>(
        y0, W1, b1, y1);

    // 8) BN stats + BN/ReLU + transpose to out [B,128,N]
    reduce_stats_kernel<H1_><<<256, H1_, 0, stream>>>(y1, sum1, sq1);
    bn_relu_out_kernel<<<dim3(N_ / 256, H1_, B_), 256, 0, stream>>>(
        y1, sum1, sq1, g1, beta1, out);
}